// TrafficGNN_27917287424795
// MI455X (gfx1250) — compile-verified
//
#include <hip/hip_runtime.h>
#include <hip/hip_bf16.h>

#define N_NODES 50000
#define N_EDGES 800000
#define NODE_IN 64
#define EDGE_IN 16
#define HIDDEN  128
#define OUT_DIM 32
#define KTOT    (HIDDEN * 2 + EDGE_IN)   // 272
#define A_PITCH (KTOT + 1)               // 273, avoids LDS bank conflicts

typedef __attribute__((ext_vector_type(2))) float v2f;
typedef __attribute__((ext_vector_type(8))) float v8f;

// D = A(16x4) * B(4x16) + C(16x16), fp32 WMMA (codegen-confirmed round 0).
__device__ __forceinline__ v8f wmma_f32_k4(v2f a, v2f b, v8f c) {
#if defined(__gfx1250__) && __has_builtin(__builtin_amdgcn_wmma_f32_16x16x4_f32)
    return __builtin_amdgcn_wmma_f32_16x16x4_f32(false, a, false, b, (short)0, c,
                                                 false, false);
#else
    c[0] += a.x * b.x + a.y * b.y;   // placeholder; real path is the builtin
    return c;
#endif
}

// ---------------- degree / normalization ----------------

__global__ void deg_init_kernel(float* __restrict__ deg) {
    int i = blockIdx.x * blockDim.x + threadIdx.x;
    if (i < N_NODES) deg[i] = 1.0f;              // self loop
}

__global__ void deg_count_kernel(float* __restrict__ deg,
                                 const int* __restrict__ eidx) {
    int e = blockIdx.x * blockDim.x + threadIdx.x;
    if (e < N_EDGES) atomicAdd(&deg[eidx[N_EDGES + e]], 1.0f);  // dst
}

__global__ void deg_finalize_kernel(float* __restrict__ deg) {
    int i = blockIdx.x * blockDim.x + threadIdx.x;
    if (i < N_NODES) deg[i] = rsqrtf(deg[i]);    // deg >= 1 always
}

// ---------------- dense fp32 WMMA GEMM: C[M,N] = A[M,K] @ B[K,N] ------------
// K, N compile-time so all strides are immediates. Block = 128 (4 waves),
// wave w owns N-tile (blockIdx.x*64 + w*16); grid = (N/64, M/16).

template <int K, int N>
__global__ void gemm_wmma_kernel(const float* __restrict__ A,
                                 const float* __restrict__ B,
                                 float* __restrict__ C) {
    const int wave = threadIdx.x >> 5;
    const int lane = threadIdx.x & 31;
    const int half = lane >> 4;
    const int l    = lane & 15;
    const int n0   = blockIdx.x * 64 + wave * 16;
    const int m0   = blockIdx.y * 16;

    const float* ap = A + (size_t)(m0 + l) * K + 2 * half;   // +4 per step
    const float* bp = B + (size_t)(2 * half) * N + n0 + l;   // +4N per step
    v8f acc = {};

#pragma unroll
    for (int k = 0; k < K; k += 4) {
        v2f a, b;
        a.x = ap[0];
        a.y = ap[1];
        b.x = bp[0];
        b.y = bp[N];
        acc = wmma_f32_k4(a, b, acc);
        ap += 4;
        bp += 4 * N;
    }

    float* crow = C + (size_t)(m0 + half * 8) * N + n0 + l;
#pragma unroll
    for (int j = 0; j < 8; ++j)
        crow[(size_t)j * N] = acc[j];
}

// ---------------- aggregation: agg = D^-1/2 (A+I) D^-1/2 h ------------------

__global__ void self_loop_init_kernel(float4* __restrict__ agg4,
                                      const float4* __restrict__ h4,
                                      const float* __restrict__ dinv) {
    int i = blockIdx.x * blockDim.x + threadIdx.x;       // over N*H/4
    if (i < N_NODES * (HIDDEN / 4)) {
        int node = i >> 5;                               // HIDDEN/4 == 32
        float d = dinv[node];
        float s = d * d;
        float4 v = h4[i];
        v.x *= s; v.y *= s; v.z *= s; v.w *= s;
        agg4[i] = v;
    }
}

// one edge handled by 32 threads, 4 floats each (HIDDEN = 128)
__global__ void scatter_edges_kernel(float* __restrict__ agg,
                                     const float* __restrict__ h,
                                     const float* __restrict__ dinv,
                                     const int* __restrict__ eidx) {
    long long tid = (long long)blockIdx.x * blockDim.x + threadIdx.x;
    if (tid >= (long long)N_EDGES * 32) return;
    int e  = (int)(tid >> 5);
    int c4 = ((int)tid & 31) * 4;
    int row = eidx[e];
    int col = eidx[N_EDGES + e];
    float norm = dinv[row] * dinv[col];
    const float4 hv = *(const float4*)(h + (size_t)row * HIDDEN + c4);
    float* ar = agg + (size_t)col * HIDDEN + c4;
    atomicAdd(ar + 0, hv.x * norm);
    atomicAdd(ar + 1, hv.y * norm);
    atomicAdd(ar + 2, hv.z * norm);
    atomicAdd(ar + 3, hv.w * norm);
}

__global__ void bias_relu_kernel(float4* __restrict__ dst4,
                                 const float4* __restrict__ agg4,
                                 const float4* __restrict__ bias4) {
    int i = blockIdx.x * blockDim.x + threadIdx.x;       // over N*H/4
    if (i < N_NODES * (HIDDEN / 4)) {
        float4 v = agg4[i];
        float4 b = bias4[i & 31];                        // HIDDEN/4 - 1
        v.x = fmaxf(v.x + b.x, 0.0f);
        v.y = fmaxf(v.y + b.y, 0.0f);
        v.z = fmaxf(v.z + b.z, 0.0f);
        v.w = fmaxf(v.w + b.w, 0.0f);
        dst4[i] = v;
    }
}

// ---------------- edge MLP: out[e] = [h[src]|h[dst]|eattr] @ We + be --------
// block = 64 threads (2 waves); block owns 16 edges; wave w computes cols
// [16w, 16w+16). Gathered 16x272 A tile staged in LDS (pitch 273).

__global__ void edge_mlp_kernel(const float* __restrict__ h,
                                const float* __restrict__ eattr,
                                const float* __restrict__ We,
                                const float* __restrict__ be,
                                const int* __restrict__ eidx,
                                float* __restrict__ out) {
    __shared__ float Atile[16 * A_PITCH];

    const int e0 = blockIdx.x * 16;

    // cooperative gather of the A tile: 4 threads per edge row
    {
        const int m   = threadIdx.x >> 2;        // 0..15
        const int sub = threadIdx.x & 3;
        const int e   = e0 + m;
        const int src = eidx[e];
        const int dst = eidx[N_EDGES + e];
        const float* hs = h + (size_t)src * HIDDEN;
        const float* hd = h + (size_t)dst * HIDDEN;
        const float* ea = eattr + (size_t)e * EDGE_IN;
        float* arow = Atile + m * A_PITCH;
#pragma unroll
        for (int k0 = 0; k0 < HIDDEN; k0 += 16) {
            float4 a = *(const float4*)(hs + k0 + sub * 4);
            float4 b = *(const float4*)(hd + k0 + sub * 4);
            *(float4*)(arow + k0 + sub * 4) = a;            // pitch 273 is odd,
            arow[HIDDEN + k0 + sub * 4 + 0] = b.x;          // but float4 into
            arow[HIDDEN + k0 + sub * 4 + 1] = b.y;          // LDS row base is
            arow[HIDDEN + k0 + sub * 4 + 2] = b.z;          // only 4B aligned:
            arow[HIDDEN + k0 + sub * 4 + 3] = b.w;          // write scalars
        }
        float4 a = *(const float4*)(ea + sub * 4);
        arow[2 * HIDDEN + sub * 4 + 0] = a.x;
        arow[2 * HIDDEN + sub * 4 + 1] = a.y;
        arow[2 * HIDDEN + sub * 4 + 2] = a.z;
        arow[2 * HIDDEN + sub * 4 + 3] = a.w;
    }
    __syncthreads();

    const int wave = threadIdx.x >> 5;           // 0 or 1
    const int lane = threadIdx.x & 31;
    const int half = lane >> 4;
    const int l    = lane & 15;
    const int n0   = wave * 16;

    const float* arow = Atile + l * A_PITCH + 2 * half;            // +4 / step
    const float* bp   = We + (size_t)(2 * half) * OUT_DIM + n0 + l; // +4*32 / step
    v8f acc = {};

#pragma unroll
    for (int k = 0; k < KTOT; k += 4) {
        v2f a, b;
        a.x = arow[0];
        a.y = arow[1];
        b.x = bp[0];
        b.y = bp[OUT_DIM];
        acc = wmma_f32_k4(a, b, acc);
        arow += 4;
        bp += 4 * OUT_DIM;
    }

    const float bias = be[n0 + l];
    float* crow = out + (size_t)(e0 + half * 8) * OUT_DIM + n0 + l;
#pragma unroll
    for (int j = 0; j < 8; ++j)
        crow[(size_t)j * OUT_DIM] = acc[j] + bias;
}

// ---------------- launch ----------------

extern "C" void kernel_launch(void* const* d_in, const int* in_sizes, int n_in,
                              void* d_out, int out_size, void* d_ws, size_t ws_size,
                              hipStream_t stream) {
    (void)in_sizes; (void)n_in; (void)out_size; (void)ws_size;

    const float* x     = (const float*)d_in[0];
    const int*   eidx  = (const int*)d_in[1];
    const float* eattr = (const float*)d_in[2];
    const float* W1    = (const float*)d_in[3];
    const float* b1    = (const float*)d_in[4];
    const float* W2    = (const float*)d_in[5];
    const float* b2    = (const float*)d_in[6];
    const float* We    = (const float*)d_in[7];
    const float* be    = (const float*)d_in[8];
    float*       out   = (float*)d_out;

    float* dinv = (float*)d_ws;
    float* bufH = dinv + ((N_NODES + 255) & ~255);        // [N, HIDDEN]
    float* bufA = bufH + (size_t)N_NODES * HIDDEN;        // [N, HIDDEN]

    const int TB = 256;
    const int gN  = (N_NODES + TB - 1) / TB;
    const int gE  = (N_EDGES + TB - 1) / TB;
    const int gV4 = (N_NODES * (HIDDEN / 4) + TB - 1) / TB;
    const int gSc = (int)(((long long)N_EDGES * 32 + TB - 1) / TB);

    // normalization coefficients
    deg_init_kernel<<<gN, TB, 0, stream>>>(dinv);
    deg_count_kernel<<<gE, TB, 0, stream>>>(dinv, eidx);
    deg_finalize_kernel<<<gN, TB, 0, stream>>>(dinv);

    // layer 1: bufH = x @ W1 ; bufA = norm-agg ; bufH = relu(bufA + b1)
    gemm_wmma_kernel<NODE_IN, HIDDEN>
        <<<dim3(HIDDEN / 64, N_NODES / 16), 128, 0, stream>>>(x, W1, bufH);
    self_loop_init_kernel<<<gV4, TB, 0, stream>>>(
        (float4*)bufA, (const float4*)bufH, dinv);
    scatter_edges_kernel<<<gSc, TB, 0, stream>>>(bufA, bufH, dinv, eidx);
    bias_relu_kernel<<<gV4, TB, 0, stream>>>(
        (float4*)bufH, (const float4*)bufA, (const float4*)b1);

    // layer 2: bufA = bufH @ W2 ; bufH = norm-agg ; bufA = relu(bufH + b2)
    gemm_wmma_kernel<HIDDEN, HIDDEN>
        <<<dim3(HIDDEN / 64, N_NODES / 16), 128, 0, stream>>>(bufH, W2, bufA);
    self_loop_init_kernel<<<gV4, TB, 0, stream>>>(
        (float4*)bufH, (const float4*)bufA, dinv);
    scatter_edges_kernel<<<gSc, TB, 0, stream>>>(bufH, bufA, dinv, eidx);
    bias_relu_kernel<<<gV4, TB, 0, stream>>>(
        (float4*)bufA, (const float4*)bufH, (const float4*)b2);

    // edge MLP over gathered features
    edge_mlp_kernel<<<N_EDGES / 16, 64, 0, stream>>>(
        bufA, eattr, We, be, eidx, out);
}